// GruDirection3d_3478923509774
// MI455X (gfx1250) — compile-verified
//
#include <hip/hip_runtime.h>

// ---------------------------------------------------------------------------
// GruDirection3d forward (d1=d2=d3=1):
//   out[d,y,x] = z*h_tilde + (1-z)*out[d-1,y-1,x-1],  border -> h0
// Memory-bound (75 MB traffic, ~25 MFLOP). One workgroup per (b,c) channel,
// slice-by-slice over d, prev slice in LDS, z/h_tilde double-buffered in LDS
// via gfx1250 async global->LDS loads (ASYNCcnt).
// ---------------------------------------------------------------------------

#if __has_builtin(__builtin_amdgcn_global_load_async_to_lds_b128) && \
    __has_builtin(__builtin_amdgcn_s_wait_asynccnt)
#define GRU_ASYNC 1
#else
#define GRU_ASYNC 0
#warning "gfx1250 async-to-LDS builtins not found; using synchronous fallback"
#endif

namespace {

constexpr int Dd = 32, Hh = 32, Ww = 32;
constexpr int SLICE = Hh * Ww;   // 1024 floats per d-slice per channel
constexpr int LPAD  = 33;        // padded prev-row stride: conflict-free shifted reads

typedef int v4i __attribute__((vector_size(16)));
typedef __attribute__((address_space(1))) v4i* gp_v4i;   // global int4*
typedef __attribute__((address_space(3))) v4i* lp_v4i;   // LDS int4*

__device__ __forceinline__ void async_copy_b128(const float* gsrc, float* ldst) {
#if GRU_ASYNC
  __builtin_amdgcn_global_load_async_to_lds_b128(
      (gp_v4i)(unsigned long long)(const void*)gsrc,  // flat & global share bit pattern
      (lp_v4i)ldst,                                    // generic -> AS(3) cast
      /*imm offset=*/0, /*cpol=*/0);
#else
  *(float4*)ldst = *(const float4*)gsrc;
#endif
}

__device__ __forceinline__ void wait_async_le2() {
#if GRU_ASYNC
  __builtin_amdgcn_s_wait_asynccnt(2);  // 2 newest (slice d+1) may remain in flight
#endif
}
__device__ __forceinline__ void wait_async_le0() {
#if GRU_ASYNC
  __builtin_amdgcn_s_wait_asynccnt(0);
#endif
}

} // namespace

__global__ __launch_bounds__(256)
void gru_dir3d_fwd(const float* __restrict__ z,
                   const float* __restrict__ ht,
                   const float* __restrict__ h0p,
                   float* __restrict__ out) {
  __shared__ float prev[Hh * LPAD];     // previous out slice (padded rows)
  __shared__ float zb[2][SLICE];        // double-buffered z slice
  __shared__ float hb[2][SLICE];        // double-buffered h_tilde slice

  const int t  = threadIdx.x;           // 0..255
  const int y  = t >> 3;                // row 0..31
  const int x0 = (t & 7) << 2;          // x-quad start 0,4,...,28
  const int loff = y * Ww + x0;

  const size_t base = (size_t)blockIdx.x * (size_t)(Dd * SLICE) + (size_t)loff;
  const float* gz = z  + base;
  const float* gh = ht + base;
  float*       go = out + base;

  const float h0 = h0p[0];

  // Prefetch slice 0 into buffer 0 (2 async ops outstanding).
  async_copy_b128(gz, &zb[0][loff]);
  async_copy_b128(gh, &hb[0][loff]);

  for (int d = 0; d < Dd; ++d) {
    const int cur = d & 1;

    // Issue slice d+1 into the other buffer, then wait for slice d only.
    // Async loads complete in order, so asynccnt<=2 => slice d is in LDS.
    if (d + 1 < Dd) {
      async_copy_b128(gz + (size_t)(d + 1) * SLICE, &zb[cur ^ 1][loff]);
      async_copy_b128(gh + (size_t)(d + 1) * SLICE, &hb[cur ^ 1][loff]);
      wait_async_le2();
    } else {
      wait_async_le0();
    }

    // Each thread reads back exactly the LDS chunk its own wave copied:
    // no cross-wave barrier needed for the staging buffers.
    const float4 z4 = *(const float4*)&zb[cur][loff];
    const float4 h4 = *(const float4*)&hb[cur][loff];

    float p0, p1, p2, p3;
    if (d == 0 || y == 0) {
      p0 = p1 = p2 = p3 = h0;
    } else {
      const float* pr = &prev[(y - 1) * LPAD + (x0 - 1)];
      p0 = (x0 == 0) ? h0 : pr[0];
      p1 = pr[1];
      p2 = pr[2];
      p3 = pr[3];
    }

    // out = z*h_tilde + (1-z)*prev  (keep reference-style evaluation order)
    float4 o4;
    o4.x = fmaf(z4.x, h4.x, (1.0f - z4.x) * p0);
    o4.y = fmaf(z4.y, h4.y, (1.0f - z4.y) * p1);
    o4.z = fmaf(z4.z, h4.z, (1.0f - z4.z) * p2);
    o4.w = fmaf(z4.w, h4.w, (1.0f - z4.w) * p3);

    *(float4*)(go + (size_t)d * SLICE) = o4;

    __syncthreads();                    // all shifted reads of prev are done
    float* pw = &prev[y * LPAD + x0];
    pw[0] = o4.x; pw[1] = o4.y; pw[2] = o4.z; pw[3] = o4.w;
    __syncthreads();                    // prev slice d visible to everyone
  }
}

extern "C" void kernel_launch(void* const* d_in, const int* in_sizes, int n_in,
                              void* d_out, int out_size, void* d_ws, size_t ws_size,
                              hipStream_t stream) {
  (void)n_in; (void)d_ws; (void)ws_size; (void)out_size;
  const float* z  = (const float*)d_in[0];
  const float* ht = (const float*)d_in[1];
  const float* h0 = (const float*)d_in[2];
  float* out = (float*)d_out;

  const int nch = in_sizes[0] / (Dd * Hh * Ww);   // B*C = 192 channels
  if (nch <= 0) return;
  gru_dir3d_fwd<<<dim3(nch), dim3(256), 0, stream>>>(z, ht, h0, out);
}